// LotoGNN_19851338842261
// MI455X (gfx1250) — compile-verified
//
#include <hip/hip_runtime.h>
#include <hip/hip_bf16.h>

typedef __attribute__((ext_vector_type(2))) float v2f;
typedef __attribute__((ext_vector_type(8))) float v8f;

#define N_NODES 100000
#define N_EDGES 3200000

// ---------------------------------------------------------------------------
// Degree count: deg[dst[e]] += 1.0f  (fp32 atomic, L2-resident)
// ---------------------------------------------------------------------------
__global__ void deg_kernel(const int* __restrict__ dst, float* __restrict__ deg, int E) {
    int e = blockIdx.x * blockDim.x + threadIdx.x;
    if (e < E) atomicAdd(&deg[dst[e]], 1.0f);
}

// deg -> rsqrt(deg + 1) in place
__global__ void dinv_kernel(float* __restrict__ deg, int N) {
    int i = blockIdx.x * blockDim.x + threadIdx.x;
    if (i < N) deg[i] = rsqrtf(deg[i] + 1.0f);
}

// ---------------------------------------------------------------------------
// C[M, NCOLS] = A[M, K] * B[K, NCOLS], fp32 via V_WMMA_F32_16X16X4_F32.
// One wave computes one 16x16 tile; block has (NCOLS/16) waves covering one
// 16-row stripe. M must be a multiple of 16 (100000 = 6250*16).
// K, NCOLS are compile-time so the K-loop fully unrolls with unconditional
// loads; the K%4 tail uses address-clamp + value-select (v_cndmask, no
// exec-mask branching, no OOB access).
// Fragment layouts per CDNA5 ISA 7.12.2:
//   A 16x4 f32: lane L holds row M=L%16; VGPR0 = K0+2*(L/16), VGPR1 = +1
//   B 4x16 f32: lane L holds col N=L%16; VGPR0 = row K0+2*(L/16), VGPR1 = +1
//   C/D 16x16:  VGPR r, lane L -> M = r + 8*(L/16), N = L%16
// ---------------------------------------------------------------------------
template <int K, int NCOLS>
__global__ void gemm_wmma_kernel(const float* __restrict__ A,
                                 const float* __restrict__ B,
                                 float* __restrict__ C) {
    const int lane  = threadIdx.x & 31;
    const int wave  = threadIdx.x >> 5;
    const int half  = lane >> 4;     // 0: lanes 0-15, 1: lanes 16-31
    const int l16   = lane & 15;
    const int mBase = blockIdx.x * 16;
    const int nBase = wave * 16;

    const size_t aRow = (size_t)(mBase + l16) * (size_t)K;
    const float* Bc   = B + nBase + l16;

    v8f acc = {};

    constexpr int KMAIN = (K / 4) * 4;
#pragma unroll
    for (int k0 = 0; k0 < KMAIN; k0 += 4) {
        const int ka = k0 + 2 * half;   // this lane's first K index
        v2f a, b;
        if constexpr ((K % 2) == 0) {
            // row stride even -> fragment is 8-byte aligned: single b64 load
            a = *reinterpret_cast<const v2f*>(A + aRow + ka);
        } else {
            a.x = A[aRow + ka];
            a.y = A[aRow + ka + 1];
        }
        b.x = Bc[(size_t)ka * NCOLS];
        b.y = Bc[(size_t)(ka + 1) * NCOLS];
        acc = __builtin_amdgcn_wmma_f32_16x16x4_f32(
            false, a, false, b, (short)0, acc, false, false);
    }

    if constexpr (KMAIN < K) {
        // Tail K-slice [KMAIN, KMAIN+4): clamp addresses in-bounds, select 0
        // for out-of-range K. Loads are unconditional (no exec branching).
        const int ka0 = KMAIN + 2 * half;
        const int ka1 = ka0 + 1;
        const int ca0 = (ka0 < K) ? ka0 : (K - 1);
        const int ca1 = (ka1 < K) ? ka1 : (K - 1);
        const float a0 = A[aRow + ca0];
        const float a1 = A[aRow + ca1];
        const float b0 = Bc[(size_t)ca0 * NCOLS];
        const float b1 = Bc[(size_t)ca1 * NCOLS];
        v2f a, b;
        a.x = (ka0 < K) ? a0 : 0.0f;
        a.y = (ka1 < K) ? a1 : 0.0f;
        b.x = (ka0 < K) ? b0 : 0.0f;
        b.y = (ka1 < K) ? b1 : 0.0f;
        acc = __builtin_amdgcn_wmma_f32_16x16x4_f32(
            false, a, false, b, (short)0, acc, false, false);
    }

    float* Cp = C + (size_t)(mBase + 8 * half) * NCOLS + nBase + l16;
#pragma unroll
    for (int r = 0; r < 8; ++r) Cp[(size_t)r * NCOLS] = acc[r];
}

// ---------------------------------------------------------------------------
// Edge scatter: AGG[dst] += H[src] * dinv[src]*dinv[dst]
// F/4 lanes per edge, float4 gathers (coalesced), fp32 atomic scatter.
// ---------------------------------------------------------------------------
template <int F>
__global__ void scatter_kernel(const float* __restrict__ H,
                               const int* __restrict__ src,
                               const int* __restrict__ dst,
                               const float* __restrict__ dinv,
                               float* __restrict__ AGG, int E) {
    constexpr int TPE = F / 4;
    const int tid = blockIdx.x * blockDim.x + threadIdx.x;
    const int e  = tid / TPE;
    const int fq = tid % TPE;
    if (e >= E) return;
    const int s = src[e];
    const int d = dst[e];
    const float coef = dinv[s] * dinv[d];
    const float4 hv = *reinterpret_cast<const float4*>(H + (size_t)s * F + fq * 4);
    float* base = AGG + (size_t)d * F + fq * 4;
    atomicAdd(base + 0, hv.x * coef);
    atomicAdd(base + 1, hv.y * coef);
    atomicAdd(base + 2, hv.z * coef);
    atomicAdd(base + 3, hv.w * coef);
}

// AGG[i,f] = relu(AGG[i,f] + H[i,f]*dinv[i]^2 + b[f])   (in place)
__global__ void selfloop_bias_relu_kernel(float* __restrict__ AGG,
                                          const float* __restrict__ H,
                                          const float* __restrict__ dinv,
                                          const float* __restrict__ b,
                                          int N, int F) {
    const int idx = blockIdx.x * blockDim.x + threadIdx.x;
    if (idx >= N * F) return;
    const int i = idx / F;
    const int f = idx % F;
    const float di = dinv[i];
    const float v = AGG[idx] + H[idx] * di * di + b[f];
    AGG[idx] = fmaxf(v, 0.0f);
}

// Fused layer-2 epilogue + FC: one wave per node (32 feats == 32 lanes),
// relu(agg + h*dinv^2 + b2) dotted with fcW via wave32 shuffle reduction.
__global__ void final_fc_kernel(const float* __restrict__ AGG2,
                                const float* __restrict__ H2,
                                const float* __restrict__ dinv,
                                const float* __restrict__ b2,
                                const float* __restrict__ fcW,
                                const float* __restrict__ fcb,
                                float* __restrict__ out, int N) {
    const int lane = threadIdx.x & 31;
    const int node = (blockIdx.x * blockDim.x + threadIdx.x) >> 5;
    if (node >= N) return;
    const float di = dinv[node];
    const size_t idx = (size_t)node * 32 + lane;
    float v = AGG2[idx] + H2[idx] * di * di + b2[lane];
    v = fmaxf(v, 0.0f) * fcW[lane];
#pragma unroll
    for (int off = 16; off > 0; off >>= 1) v += __shfl_xor(v, off, 32);
    if (lane == 0) out[node] = v + fcb[0];
}

// ---------------------------------------------------------------------------
extern "C" void kernel_launch(void* const* d_in, const int* in_sizes, int n_in,
                              void* d_out, int out_size, void* d_ws, size_t ws_size,
                              hipStream_t stream) {
    const float* x    = (const float*)d_in[0];            // [N, 37]
    const int*   eidx = (const int*)d_in[1];              // [2, E]
    const float* W1   = (const float*)d_in[2];            // [37, 64]
    const float* b1   = (const float*)d_in[3];            // [64]
    const float* W2   = (const float*)d_in[4];            // [64, 32]
    const float* b2   = (const float*)d_in[5];            // [32]
    const float* fcW  = (const float*)d_in[6];            // [32]
    const float* fcb  = (const float*)d_in[7];            // [1]
    float* out = (float*)d_out;                           // [N]

    const int* src = eidx;
    const int* dst = eidx + N_EDGES;

    // Workspace layout (byte offsets, MB-aligned):
    char* ws = (char*)d_ws;
    float* dinv = (float*)(ws + 0);                        //  0.4 MB
    float* h1   = (float*)(ws + (size_t)1  * 1024 * 1024); // 25.6 MB [N,64]
    float* agg1 = (float*)(ws + (size_t)27 * 1024 * 1024); // 25.6 MB [N,64]
    float* h2   = (float*)(ws + (size_t)53 * 1024 * 1024); // 12.8 MB [N,32]
    float* agg2 = (float*)(ws + (size_t)66 * 1024 * 1024); // 12.8 MB [N,32]

    // --- degree / normalization ---
    hipMemsetAsync(dinv, 0, (size_t)N_NODES * sizeof(float), stream);
    deg_kernel<<<(N_EDGES + 255) / 256, 256, 0, stream>>>(dst, dinv, N_EDGES);
    dinv_kernel<<<(N_NODES + 255) / 256, 256, 0, stream>>>(dinv, N_NODES);

    // --- layer 1: h1 = x @ W1  (WMMA fp32, K=37 w/ clamped tail, N=64) ---
    gemm_wmma_kernel<37, 64><<<N_NODES / 16, (64 / 16) * 32, 0, stream>>>(x, W1, h1);

    hipMemsetAsync(agg1, 0, (size_t)N_NODES * 64 * sizeof(float), stream);
    {
        const long long threads = (long long)N_EDGES * (64 / 4);
        scatter_kernel<64><<<(int)((threads + 255) / 256), 256, 0, stream>>>(
            h1, src, dst, dinv, agg1, N_EDGES);
    }
    selfloop_bias_relu_kernel<<<(N_NODES * 64 + 255) / 256, 256, 0, stream>>>(
        agg1, h1, dinv, b1, N_NODES, 64);

    // --- layer 2: h2 = relu(agg1) @ W2  (WMMA fp32, K=64, N=32) ---
    gemm_wmma_kernel<64, 32><<<N_NODES / 16, (32 / 16) * 32, 0, stream>>>(agg1, W2, h2);

    hipMemsetAsync(agg2, 0, (size_t)N_NODES * 32 * sizeof(float), stream);
    {
        const long long threads = (long long)N_EDGES * (32 / 4);
        scatter_kernel<32><<<(int)((threads + 255) / 256), 256, 0, stream>>>(
            h2, src, dst, dinv, agg2, N_EDGES);
    }

    // --- fused epilogue + FC: one wave per node ---
    final_fc_kernel<<<(N_NODES * 32 + 255) / 256, 256, 0, stream>>>(
        agg2, h2, dinv, b2, fcW, fcb, out, N_NODES);
}